// ContextKnowledgeEncoder_36103495090441
// MI455X (gfx1250) — compile-verified
//
#include <hip/hip_runtime.h>
#include <hip/hip_bf16.h>
#include <math.h>

// Problem constants from the reference: N=8, K=32, Tk=128, Tc=256, V=32000, D=512
#define N_   8
#define K_   32
#define TK_  128
#define TC_  256
#define D_   512
#define D4_  (D_ / 4)          // 128 float4 per embedding row
#define TFULL_ (TK_ + TC_)     // 384
#define NEGINF_F (-1e20f)
#define INV_SQRT_D 0.044194173824159216f  // 1/sqrt(512)
#define PF_DIST 8              // token-row prefetch distance

typedef __attribute__((ext_vector_type(2))) float v2f;
typedef __attribute__((ext_vector_type(8))) float v8f;

// ---------------------------------------------------------------------------
// Kernel 1: universal-sentence-embedding pooling (gather + masked sum / sqrt(len)).
// One block (128 threads = 4 waves) per pooled row; each thread owns one
// float4 column slice -> each token's 2KB embedding row is one fully
// coalesced block-wide load. Branch-free masked accumulate (load emb[tok]
// unconditionally, multiply by 0/1 mask): pad tokens are ~0.003% of a
// 32000-vocab uniform draw, and removing the divergent branch lets the
// compiler unroll and keep multiple global_load_b128 gathers in flight per
// wave (latency hiding for the 64MB gather that dominates the roofline).
// Deep prefetch of the row PF_DIST tokens ahead emits global_prefetch_b8.
// ---------------------------------------------------------------------------
__global__ void __launch_bounds__(128)
use_pool_kernel(const int* __restrict__ tokens,   // [R, T]
                int T,
                const float* __restrict__ emb,    // [V, D]
                float* __restrict__ out)          // [R, D] = sum/sqrt(cnt) * 1/sqrt(D)
{
    __shared__ int toks[TC_];                     // T <= 256
    const int r   = blockIdx.x;
    const int tid = threadIdx.x;                  // 0..127

    for (int t = tid; t < T; t += 128) toks[t] = tokens[r * T + t];
    __syncthreads();

    const float4* __restrict__ emb4 = (const float4*)emb;
    float4 acc = make_float4(0.f, 0.f, 0.f, 0.f);
    int cnt = 0;

    #pragma unroll 4
    for (int t = 0; t < T; ++t) {
        const int tok = toks[t];
        if (t + PF_DIST < T)
            __builtin_prefetch(
                (const void*)(emb4 + (size_t)toks[t + PF_DIST] * D4_ + tid), 0, 0);
        const float m = (tok != 0) ? 1.0f : 0.0f;     // mask = token != pad(0)
        cnt += (tok != 0);
        const float4 e = emb4[(size_t)tok * D4_ + tid];
        acc.x = fmaf(e.x, m, acc.x);
        acc.y = fmaf(e.y, m, acc.y);
        acc.z = fmaf(e.z, m, acc.z);
        acc.w = fmaf(e.w, m, acc.w);
    }

    // scale = (1/sqrt(cnt)) * (1/sqrt(D)); cnt==0 -> inf/nan like the reference's 0-div
    const float scale = INV_SQRT_D / sqrtf((float)cnt);
    acc.x *= scale; acc.y *= scale; acc.z *= scale; acc.w *= scale;
    ((float4*)out)[(size_t)r * D4_ + tid] = acc;
}

// ---------------------------------------------------------------------------
// Kernel 2: ck_attn[n,k] = know_use[n,k,:] . context_use[n,:] via
// V_WMMA_F32_16X16X4_F32 (fp32 matrix core path, exact arithmetic).
// 16 waves; wave w handles n = w/2, k-tile = (w%2)*16.
//   A (16x4): A[m][kk] = know_use[n, k0+m, d0+kk]
//     lanes 0-15: m=lane, VGPR0=K0, VGPR1=K1 ; lanes 16-31: m=lane-16, K2/K3
//   B (4x16): every column = context_use[n, d0..d0+3]  (same value all 16 cols)
//   C/D (16x16 f32, 8 VGPRs): column N=0 lives in lane 0 (M=0..7 in c[0..7])
//     and lane 16 (M=8..15 in c[0..7]).
// Then mask with ck_mask -> NEGINF, argmax per n (first-max like jnp.argmax),
// honor use_cs_ids.
// ---------------------------------------------------------------------------
__global__ void __launch_bounds__(512)
ck_attn_kernel(const float* __restrict__ know_use,     // [N,K,D]
               const float* __restrict__ context_use,  // [N,D]
               const int*   __restrict__ ck_mask,      // [N,K]
               const int*   __restrict__ cs_ids_in,    // [N]
               const int*   __restrict__ use_cs_ids,   // [1]
               float*       __restrict__ ck_attn_out,  // [N,K] (masked)
               int*         __restrict__ cs_ids_sel)   // [N] (workspace)
{
    __shared__ float ck_s[N_ * K_];

    const int tid  = threadIdx.x;       // 0..511
    const int wave = tid >> 5;          // 0..15
    const int lane = tid & 31;
    const int n    = wave >> 1;
    const int k0   = (wave & 1) << 4;   // 0 or 16
    const int m    = lane & 15;
    const int koff = (lane < 16) ? 0 : 2;

    const float* __restrict__ arow = know_use    + (size_t)(n * K_ + k0 + m) * D_;
    const float* __restrict__ brow = context_use + (size_t)n * D_;

    v8f c = {0.f, 0.f, 0.f, 0.f, 0.f, 0.f, 0.f, 0.f};

    // Accumulate over D=512 in K=4 steps: 128 WMMAs per wave (uniform flow, EXEC all 1s)
    for (int d0 = 0; d0 < D_; d0 += 4) {
        v2f a, b;
        a.x = arow[d0 + koff];
        a.y = arow[d0 + koff + 1];
        b.x = brow[d0 + koff];
        b.y = brow[d0 + koff + 1];
        c = __builtin_amdgcn_wmma_f32_16x16x4_f32(
                /*neg_a=*/false, a, /*neg_b=*/false, b,
                /*c_mod=*/(short)0, c, /*reuse_a=*/false, /*reuse_b=*/false);
    }

    // Extract column N=0 of the 16x16 D tile.
    if (lane == 0) {
        #pragma unroll
        for (int j = 0; j < 8; ++j) ck_s[n * K_ + k0 + j] = c[j];
    } else if (lane == 16) {
        #pragma unroll
        for (int j = 0; j < 8; ++j) ck_s[n * K_ + k0 + 8 + j] = c[j];
    }
    __syncthreads();

    if (tid < N_) {
        const int nn = tid;
        float best = -INFINITY;
        int besti = 0;
        for (int k = 0; k < K_; ++k) {
            const float v  = ck_s[nn * K_ + k];
            const float mv = (ck_mask[nn * K_ + k] != 0) ? v : NEGINF_F;
            ck_attn_out[nn * K_ + k] = mv;
            if (mv > best) { best = mv; besti = k; }   // strict > -> first max (jnp.argmax)
        }
        cs_ids_sel[nn] = (use_cs_ids[0] != 0) ? cs_ids_in[nn] : besti;
    }
}

// ---------------------------------------------------------------------------
// Kernel 3: write full_enc = concat(emb[know_tokens[n, cs_id]], emb[src_tokens])
// and full_mask. One block (128 threads) per output row; coalesced float4
// gather of the 2KB embedding row. Note: the reference embeds pad tokens too
// (emb[0] row), masking is carried separately -> copy unconditionally.
// ---------------------------------------------------------------------------
__global__ void __launch_bounds__(128)
writeout_kernel(const int*   __restrict__ src_tokens,   // [N,Tc]
                const int*   __restrict__ know_tokens,  // [N,K,Tk]
                const int*   __restrict__ cs_ids_sel,   // [N]
                const float* __restrict__ emb,          // [V,D]
                float*       __restrict__ full_enc,     // [N,Tfull,D]
                float*       __restrict__ full_mask)    // [N,Tfull]
{
    const int b   = blockIdx.x;          // 0 .. N*TFULL_-1
    const int n   = b / TFULL_;
    const int r   = b % TFULL_;
    const int tid = threadIdx.x;         // 0..127

    int tok;
    if (r < TK_) {
        const int cs = cs_ids_sel[n];
        tok = know_tokens[((size_t)n * K_ + cs) * TK_ + r];
    } else {
        tok = src_tokens[n * TC_ + (r - TK_)];
    }

    const float4* __restrict__ src = (const float4*)emb + (size_t)tok * D4_;
    float4* __restrict__ dst = (float4*)full_enc + ((size_t)n * TFULL_ + r) * D4_;
    dst[tid] = src[tid];

    if (tid == 0)
        full_mask[n * TFULL_ + r] = (tok != 0) ? 1.0f : 0.0f;
}

// ---------------------------------------------------------------------------
// Host-side launcher. Inputs in setup_inputs() order:
//   0 src_tokens [N*Tc] int, 1 know_tokens [N*K*Tk] int, 2 ck_mask [N*K] int,
//   3 cs_ids [N] int, 4 use_cs_ids [1] int, 5 emb [V*D] float.
// d_out (float) = full_enc | full_mask | ck_attn flat in return order.
// ---------------------------------------------------------------------------
extern "C" void kernel_launch(void* const* d_in, const int* in_sizes, int n_in,
                              void* d_out, int out_size, void* d_ws, size_t ws_size,
                              hipStream_t stream) {
    const int*   src_tokens  = (const int*)d_in[0];
    const int*   know_tokens = (const int*)d_in[1];
    const int*   ck_mask     = (const int*)d_in[2];
    const int*   cs_ids      = (const int*)d_in[3];
    const int*   use_cs_ids  = (const int*)d_in[4];
    const float* emb         = (const float*)d_in[5];

    float* out       = (float*)d_out;
    float* full_enc  = out;                                      // N*Tfull*D
    float* full_mask = out + (size_t)N_ * TFULL_ * D_;           // N*Tfull
    float* ck_attn   = full_mask + (size_t)N_ * TFULL_;          // N*K

    // Workspace: context_use [N,D] | know_use [N,K,D] | cs_ids_sel [N]
    float* context_use = (float*)d_ws;
    float* know_use    = context_use + (size_t)N_ * D_;
    int*   cs_ids_sel  = (int*)(know_use + (size_t)N_ * K_ * D_);

    // 1) pool context: 8 rows of 256 tokens         (~4 MB gather)
    use_pool_kernel<<<N_, 128, 0, stream>>>(src_tokens, TC_, emb, context_use);
    // 2) pool knowledge: 256 rows of 128 tokens     (~64 MB gather, dominant)
    use_pool_kernel<<<N_ * K_, 128, 0, stream>>>(know_tokens, TK_, emb, know_use);
    // 3) attention scores via fp32 WMMA + mask + argmax + cs_id select
    ck_attn_kernel<<<1, 512, 0, stream>>>(know_use, context_use, ck_mask,
                                          cs_ids, use_cs_ids, ck_attn, cs_ids_sel);
    // 4) gather chosen knowledge + context rows into full_enc, emit full_mask
    writeout_kernel<<<N_ * TFULL_, 128, 0, stream>>>(src_tokens, know_tokens,
                                                     cs_ids_sel, emb,
                                                     full_enc, full_mask);
}